// GCN_55138790146121
// MI455X (gfx1250) — compile-verified
//
#include <hip/hip_runtime.h>
#include <hip/hip_bf16.h>

typedef __attribute__((ext_vector_type(2))) float v2f;
typedef __attribute__((ext_vector_type(8))) float v8f;

#define WAVES_PER_BLOCK 8
#define TPB (WAVES_PER_BLOCK * 32)

__device__ __forceinline__ v8f wmma_f32_16x16x4(v2f a, v2f b, v8f c) {
    // D = A(16x4,f32) * B(4x16,f32) + C(16x16,f32)
    return __builtin_amdgcn_wmma_f32_16x16x4_f32(
        /*neg_a=*/false, a, /*neg_b=*/false, b,
        /*c_mod=*/(short)0, c, /*reuse_a=*/false, /*reuse_b=*/false);
}

// ---------------- degree / normalization ----------------
__global__ void k_deg_init(float* __restrict__ deg, int n) {
    int i = blockIdx.x * blockDim.x + threadIdx.x;
    if (i < n) deg[i] = 1.0f;  // self-loop
}

__global__ void k_deg_accum(const int* __restrict__ dst, float* __restrict__ deg, int e) {
    int i = blockIdx.x * blockDim.x + threadIdx.x;
    if (i < e) atomicAdd(&deg[dst[i]], 1.0f);
}

__global__ void k_deg_rsqrt(float* __restrict__ deg, int n) {
    int i = blockIdx.x * blockDim.x + threadIdx.x;
    if (i < n) deg[i] = rsqrtf(deg[i]);
}

// ---------------- fused MLP 131 -> 64 -> 32 -> 16 (WMMA f32) ----------------
__global__ __launch_bounds__(TPB) void k_mlp(
    const float* __restrict__ x,
    const float* __restrict__ W1, const float* __restrict__ b1,
    const float* __restrict__ W2, const float* __restrict__ b2,
    const float* __restrict__ W3, const float* __restrict__ b3,
    float* __restrict__ h0, int n)
{
    __shared__ float w1s[132 * 64];                 // W1 zero-padded to K=132
    __shared__ float s1[WAVES_PER_BLOCK][16 * 64];  // layer-1 activations
    __shared__ float s2[WAVES_PER_BLOCK][16 * 32];  // layer-2 activations

    // cooperative load of padded W1 (rows 131 zeroed)
    for (int idx = threadIdx.x; idx < 132 * 64; idx += TPB)
        w1s[idx] = (idx < 131 * 64) ? W1[idx] : 0.0f;
    __syncthreads();

    const int wave = threadIdx.x >> 5;
    const int lane = threadIdx.x & 31;
    const int half = lane >> 4;   // 0: K pair {k,k+1}, 1: {k+2,k+3}
    const int lr   = lane & 15;   // A row / B,C column
    const int tile = blockIdx.x * WAVES_PER_BLOCK + wave;
    const int m0   = tile * 16;

    int arow = m0 + lr;
    if (arow > n - 1) arow = n - 1;
    const float* xr = x + (size_t)arow * 131;

    // ---- layer 1: 131 -> 64 (K padded to 132; B rows >=131 are zero) ----
    v8f acc1[4];
#pragma unroll
    for (int t = 0; t < 4; ++t) acc1[t] = {};
#pragma unroll
    for (int k = 0; k < 132; k += 4) {
        const int k0 = k + 2 * half;            // <= 130
        const int k1 = k0 + 1;                  // <= 131 (only last iter, half=1)
        v2f a;
        a.x = xr[k0];
        a.y = xr[k1 < 131 ? k1 : 130];          // branch-free clamp; * zero B row
#pragma unroll
        for (int t = 0; t < 4; ++t) {
            v2f b;
            b.x = w1s[k0 * 64 + t * 16 + lr];
            b.y = w1s[k1 * 64 + t * 16 + lr];   // row 131 is zero-padded
            acc1[t] = wmma_f32_16x16x4(a, b, acc1[t]);
        }
    }
#pragma unroll
    for (int t = 0; t < 4; ++t)
#pragma unroll
        for (int v = 0; v < 8; ++v)
            s1[wave][(v + 8 * half) * 64 + t * 16 + lr] = tanhf(acc1[t][v] + b1[t * 16 + lr]);
    __syncthreads();

    // ---- layer 2: 64 -> 32 ----
    v8f acc2[2];
#pragma unroll
    for (int t = 0; t < 2; ++t) acc2[t] = {};
#pragma unroll
    for (int k = 0; k < 64; k += 4) {
        const int k0 = k + 2 * half;
        v2f a;
        a.x = s1[wave][lr * 64 + k0];
        a.y = s1[wave][lr * 64 + k0 + 1];
#pragma unroll
        for (int t = 0; t < 2; ++t) {
            v2f b;
            b.x = W2[k0 * 32 + t * 16 + lr];
            b.y = W2[(k0 + 1) * 32 + t * 16 + lr];
            acc2[t] = wmma_f32_16x16x4(a, b, acc2[t]);
        }
    }
#pragma unroll
    for (int t = 0; t < 2; ++t)
#pragma unroll
        for (int v = 0; v < 8; ++v)
            s2[wave][(v + 8 * half) * 32 + t * 16 + lr] = tanhf(acc2[t][v] + b2[t * 16 + lr]);
    __syncthreads();

    // ---- layer 3: 32 -> 16 (no tanh) ----
    v8f acc3 = {};
#pragma unroll
    for (int k = 0; k < 32; k += 4) {
        const int k0 = k + 2 * half;
        v2f a, b;
        a.x = s2[wave][lr * 32 + k0];
        a.y = s2[wave][lr * 32 + k0 + 1];
        b.x = W3[k0 * 16 + lr];
        b.y = W3[(k0 + 1) * 16 + lr];
        acc3 = wmma_f32_16x16x4(a, b, acc3);
    }
#pragma unroll
    for (int v = 0; v < 8; ++v) {
        int r = m0 + v + 8 * half;
        if (r < n) h0[r * 16 + lr] = acc3[v] + b3[lr];
    }
}

// ---------------- GCN transform: ht = (h @ W) * dinv; seed agg with self-loop ----
// Called with h == agg (in-place): each tile reads only its own 16 rows before
// writing them, and all loads complete (feed the WMMA chain) before any store
// issues, so no __restrict__ on h/agg.
__global__ __launch_bounds__(TPB) void k_conv_transform(
    const float* h, const float* __restrict__ W,
    const float* __restrict__ dinv,
    float* __restrict__ ht, float* agg, int n)
{
    const int wave = threadIdx.x >> 5;
    const int lane = threadIdx.x & 31;
    const int half = lane >> 4;
    const int lr   = lane & 15;
    const int tile = blockIdx.x * WAVES_PER_BLOCK + wave;
    const int m0   = tile * 16;

    int arow = m0 + lr;
    if (arow > n - 1) arow = n - 1;
    const float* hr = h + (size_t)arow * 16;

    v8f acc = {};
#pragma unroll
    for (int k = 0; k < 16; k += 4) {
        const int k0 = k + 2 * half;
        v2f a, b;
        a.x = hr[k0];
        a.y = hr[k0 + 1];
        b.x = W[k0 * 16 + lr];
        b.y = W[(k0 + 1) * 16 + lr];
        acc = wmma_f32_16x16x4(a, b, acc);
    }
#pragma unroll
    for (int v = 0; v < 8; ++v) {
        int r = m0 + v + 8 * half;
        if (r < n) {
            float val = acc[v] * dinv[r];
            ht[r * 16 + lr]  = val;  // gather source for edges
            agg[r * 16 + lr] = val;  // self-loop seed (src == dst contributes ht[i])
        }
    }
}

// ------- edge scatter: agg[dst] += ht[src]; 4 threads per edge (4 feats each) -------
__global__ void k_edge_scatter(const int* __restrict__ src, const int* __restrict__ dst,
                               const float* __restrict__ ht, float* __restrict__ agg, int e)
{
    int i = blockIdx.x * blockDim.x + threadIdx.x;
    const int edge = i >> 2;
    const int part = i & 3;
    if (edge >= e) return;
    const int s = src[edge];
    const int d = dst[edge];
    const float4 v = *(const float4*)(ht + (size_t)s * 16 + part * 4);
    float* ap = agg + (size_t)d * 16 + part * 4;
    atomicAdd(ap + 0, v.x);
    atomicAdd(ap + 1, v.y);
    atomicAdd(ap + 2, v.z);
    atomicAdd(ap + 3, v.w);
}

// ---------------- finalize: h_out = tanh(dinv[node] * agg + bias) ------------------
__global__ void k_finalize(const float* __restrict__ agg, const float* __restrict__ dinv,
                           const float* __restrict__ b, float* __restrict__ hout, int n)
{
    int i = blockIdx.x * blockDim.x + threadIdx.x;
    if (i >= n * 16) return;
    const int node = i >> 4;
    const int f    = i & 15;
    hout[i] = tanhf(dinv[node] * agg[i] + b[f]);
}

// ---------------- classifier 16 -> 2 ----------------
__global__ void k_classifier(const float* __restrict__ h, const float* __restrict__ W,
                             const float* __restrict__ b, float* __restrict__ out, int n)
{
    int i = blockIdx.x * blockDim.x + threadIdx.x;
    if (i >= n) return;
    const float* hr = h + (size_t)i * 16;
    float o0 = b[0], o1 = b[1];
#pragma unroll
    for (int f = 0; f < 16; ++f) {
        const float hv = hr[f];
        o0 += hv * W[f * 2 + 0];
        o1 += hv * W[f * 2 + 1];
    }
    out[i * 2 + 0] = o0;
    out[i * 2 + 1] = o1;
}

extern "C" void kernel_launch(void* const* d_in, const int* in_sizes, int n_in,
                              void* d_out, int out_size, void* d_ws, size_t ws_size,
                              hipStream_t stream)
{
    const float* x    = (const float*)d_in[0];
    const int*   ei   = (const int*)d_in[1];
    const float* W1   = (const float*)d_in[2];
    const float* b1   = (const float*)d_in[3];
    const float* W2   = (const float*)d_in[4];
    const float* b2   = (const float*)d_in[5];
    const float* W3   = (const float*)d_in[6];
    const float* b3   = (const float*)d_in[7];
    const float* Wc1  = (const float*)d_in[8];
    const float* bc1  = (const float*)d_in[9];
    const float* Wg   = (const float*)d_in[10];
    const float* bg   = (const float*)d_in[11];
    const float* Wcls = (const float*)d_in[12];
    const float* bcls = (const float*)d_in[13];

    const int n = in_sizes[0] / 131;
    const int e = in_sizes[1] / 2;
    const int* src = ei;       // edge_index[0]
    const int* dst = ei + e;   // edge_index[1]

    // workspace: dinv[n] | F (features/agg, 16n) | T (ht, 16n)  ~= 13 MB
    float* ws   = (float*)d_ws;
    float* dinv = ws;                       ws += n;
    float* F    = ws;                       ws += (size_t)n * 16;
    float* T    = ws;                       /* ws += 16n */

    float* outp       = (float*)d_out;
    float* hout_final = outp + (size_t)n * 2;  // second output: final h [n,16]

    const int tpb   = 256;
    const int nb_n  = (n + tpb - 1) / tpb;
    const int nb_e  = (e + tpb - 1) / tpb;
    const int nb_e4 = (e * 4 + tpb - 1) / tpb;     // 4 threads per edge
    const int nb_nf = (n * 16 + tpb - 1) / tpb;
    const int tiles = (n + 15) / 16;
    const int nb_wm = (tiles + WAVES_PER_BLOCK - 1) / WAVES_PER_BLOCK;

    // degrees -> dinv = rsqrt(deg), deg includes self-loop
    k_deg_init <<<nb_n, tpb, 0, stream>>>(dinv, n);
    k_deg_accum<<<nb_e, tpb, 0, stream>>>(dst, dinv, e);
    k_deg_rsqrt<<<nb_n, tpb, 0, stream>>>(dinv, n);

    // MLP -> F
    k_mlp<<<nb_wm, TPB, 0, stream>>>(x, W1, b1, W2, b2, W3, b3, F, n);

    // 10 conv layers; transform is per-tile in-place safe on F
    for (int layer = 0; layer < 10; ++layer) {
        const float* W = (layer < 5) ? Wc1 : (Wg + (size_t)(layer - 5) * 256);
        const float* b = (layer < 5) ? bc1 : (bg + (size_t)(layer - 5) * 16);
        float* hout = (layer == 9) ? hout_final : F;
        k_conv_transform<<<nb_wm, TPB, 0, stream>>>(F, W, dinv, T, F, n);
        k_edge_scatter  <<<nb_e4, tpb, 0, stream>>>(src, dst, T, F, e);
        k_finalize      <<<nb_nf, tpb, 0, stream>>>(F, dinv, b, hout, n);
    }

    // classifier on final h (already in d_out's h region)
    k_classifier<<<nb_n, tpb, 0, stream>>>(hout_final, Wcls, bcls, outp, n);
}